// RZGate_549755814190
// MI455X (gfx1250) — compile-verified
//
#include <hip/hip_runtime.h>
#include <hip/hip_bf16.h>

// out[n,b] = exp(i*theta_n) * x[n,b], theta_n = -0.5 * sum_i angle[i]*sdiag[bit(n,i)]
// x: (4096, 256) f32 row-major; out: (4096, 256) complex64 -> interleaved f32 pairs.
// Two-phase plan: (A) build 4096-entry {cos,sin} table (32 KB, L2-resident),
// (B) HBM-streaming apply kernel using the CDNA5 async global->LDS data mover.

#define NROWS  4096
#define NCOLS  256
#define NSITES 12
#define RPB    4     // rows per block  (tile = 4*256 floats = 4 KB)
#define TPB    256   // threads per block (8 waves on wave32)

#if defined(__gfx1250__) && __has_builtin(__builtin_amdgcn_global_load_async_to_lds_b128)
#define HAVE_ASYNC_LDS 1
#else
#define HAVE_ASYNC_LDS 0
#endif

typedef __attribute__((ext_vector_type(4))) int v4i_t;

__device__ __forceinline__ float row_theta(int row, const float* __restrict__ angle,
                                           float s0, float s1)
{
    float th = 0.0f;
#pragma unroll
    for (int i = 0; i < NSITES; ++i) {
        const int bit = (row >> (NSITES - 1 - i)) & 1;  // site 0 = MSB (kron order)
        th = fmaf(angle[i], bit ? s1 : s0, th);
    }
    return th * -0.5f;
}

// ---- Kernel A: phase table, one thread per row (runs once, negligible cost) ----
__global__ __launch_bounds__(TPB) void rz_phase_kernel(
    const float* __restrict__ angle,
    const float* __restrict__ S,
    float2* __restrict__ phase)
{
    const int row = blockIdx.x * TPB + threadIdx.x;
    const float th = row_theta(row, angle, S[0], S[3]);
    float sn, cs;
    __sincosf(th, &sn, &cs);
    phase[row] = make_float2(cs, sn);
}

// ---- Kernel B: bandwidth kernel, async-LDS staging + per-row complex scale ----
__global__ __launch_bounds__(TPB) void rz_apply_kernel(
    const float* __restrict__ x,
    const float2* __restrict__ phase,
    float* __restrict__ out)
{
    __shared__ __align__(16) float tile[RPB * NCOLS];

    const int t   = threadIdx.x;
    const int blk = blockIdx.x;

    const float* gsrc = x + (size_t)blk * (RPB * NCOLS) + t * 4;

#if HAVE_ASYNC_LDS
    __builtin_amdgcn_global_load_async_to_lds_b128(
        (v4i_t __attribute__((address_space(1)))*)gsrc,
        (v4i_t __attribute__((address_space(3)))*)&tile[t * 4],
        /*offset=*/0, /*cpol=*/0);
#else
    *(float4*)&tile[t * 4] = *(const float4*)gsrc;
#endif

    // Overlap with the in-flight copy: 8 B phase fetch (L2 hit, 64-lane broadcast).
    const int    row = blk * RPB + (t >> 6);
    const float2 p   = phase[row];

#if HAVE_ASYNC_LDS
#if __has_builtin(__builtin_amdgcn_s_wait_asynccnt)
    __builtin_amdgcn_s_wait_asynccnt(0);
#else
    asm volatile("s_wait_asynccnt 0" ::: "memory");
#endif
#endif
    __syncthreads();

    const float4 v   = *(const float4*)&tile[t * 4];
    const int    col = (t & 63) * 4;
    float*       o   = out + ((size_t)row * NCOLS + col) * 2;

    float4 o0, o1;
    o0.x = v.x * p.x; o0.y = v.x * p.y;
    o0.z = v.y * p.x; o0.w = v.y * p.y;
    o1.x = v.z * p.x; o1.y = v.z * p.y;
    o1.z = v.w * p.x; o1.w = v.w * p.y;
    *(float4*)(o)     = o0;   // 32B interleaved complex out per lane, coalesced
    *(float4*)(o + 4) = o1;
}

// ---- Fallback: fused single kernel (if scratch is too small for the table) ----
__global__ __launch_bounds__(TPB) void rz_fused_kernel(
    const float* __restrict__ x,
    const float* __restrict__ angle,
    const float* __restrict__ S,
    float* __restrict__ out)
{
    __shared__ __align__(16) float tile[RPB * NCOLS];

    const int t   = threadIdx.x;
    const int blk = blockIdx.x;
    const float* gsrc = x + (size_t)blk * (RPB * NCOLS) + t * 4;

#if HAVE_ASYNC_LDS
    __builtin_amdgcn_global_load_async_to_lds_b128(
        (v4i_t __attribute__((address_space(1)))*)gsrc,
        (v4i_t __attribute__((address_space(3)))*)&tile[t * 4], 0, 0);
#else
    *(float4*)&tile[t * 4] = *(const float4*)gsrc;
#endif

    const int   row = blk * RPB + (t >> 6);
    const float th  = row_theta(row, angle, S[0], S[3]);
    float sn, cs;
    __sincosf(th, &sn, &cs);

#if HAVE_ASYNC_LDS
#if __has_builtin(__builtin_amdgcn_s_wait_asynccnt)
    __builtin_amdgcn_s_wait_asynccnt(0);
#else
    asm volatile("s_wait_asynccnt 0" ::: "memory");
#endif
#endif
    __syncthreads();

    const float4 v   = *(const float4*)&tile[t * 4];
    const int    col = (t & 63) * 4;
    float*       o   = out + ((size_t)row * NCOLS + col) * 2;

    float4 o0, o1;
    o0.x = v.x * cs; o0.y = v.x * sn;
    o0.z = v.y * cs; o0.w = v.y * sn;
    o1.x = v.z * cs; o1.y = v.z * sn;
    o1.z = v.w * cs; o1.w = v.w * sn;
    *(float4*)(o)     = o0;
    *(float4*)(o + 4) = o1;
}

extern "C" void kernel_launch(void* const* d_in, const int* in_sizes, int n_in,
                              void* d_out, int out_size, void* d_ws, size_t ws_size,
                              hipStream_t stream)
{
    (void)in_sizes; (void)n_in; (void)out_size;
    const float* x     = (const float*)d_in[0];
    const float* angle = (const float*)d_in[1];
    const float* S     = (const float*)d_in[2];
    float*       out   = (float*)d_out;

    if (ws_size >= NROWS * sizeof(float2)) {
        float2* phase = (float2*)d_ws;
        rz_phase_kernel<<<dim3(NROWS / TPB), dim3(TPB), 0, stream>>>(angle, S, phase);
        rz_apply_kernel<<<dim3(NROWS / RPB), dim3(TPB), 0, stream>>>(x, phase, out);
    } else {
        rz_fused_kernel<<<dim3(NROWS / RPB), dim3(TPB), 0, stream>>>(x, angle, S, out);
    }
}